// GraphNCA_28183575396996
// MI455X (gfx1250) — compile-verified
//
#include <hip/hip_runtime.h>

#define N_NODES 50000
#define N_EDGES 800000
#define C3 288            // 3*C
#define HID 32

typedef __attribute__((ext_vector_type(2))) float v2f;
typedef __attribute__((ext_vector_type(8))) float v8f;

// D = A(16x4) * B(4x16) + C(16x16), fp32 WMMA (CDNA5 wave32)
#define WMMA_F32_16x16x4(a, b, c) \
    __builtin_amdgcn_wmma_f32_16x16x4_f32(false, (a), false, (b), (short)0, (c), false, false)

// ---------------------------------------------------------------------------
// Kernel 1: deg = 1.0 (self loop), features = 0
// ---------------------------------------------------------------------------
__global__ void k_init(float* __restrict__ deg, float4* __restrict__ feat4) {
    const int tid = blockIdx.x * blockDim.x + threadIdx.x;
    const int FEAT4 = N_NODES * C3 / 4;      // 3,600,000
    if (tid < FEAT4) feat4[tid] = make_float4(0.f, 0.f, 0.f, 0.f);
    if (tid < N_NODES) deg[tid] = 1.0f;
}

// ---------------------------------------------------------------------------
// Kernel 2: deg[dst] += 1 per edge
// ---------------------------------------------------------------------------
__global__ void k_degree(const int* __restrict__ ei, float* __restrict__ deg) {
    const int e = blockIdx.x * blockDim.x + threadIdx.x;
    if (e < N_EDGES) atomicAdd(&deg[ei[N_EDGES + e]], 1.0f);
}

// ---------------------------------------------------------------------------
// Kernel 3: dinv = rsqrt(deg)   (deg >= 1 always, so no zero guard needed)
// ---------------------------------------------------------------------------
__global__ void k_dinv(float* __restrict__ deg) {
    const int i = blockIdx.x * blockDim.x + threadIdx.x;
    if (i < N_NODES) deg[i] = rsqrtf(deg[i]);
}

// ---------------------------------------------------------------------------
// Kernel 4: h = x @ W_gcn   [50000,96] x [96,288]
// One wave per 16x16 output tile; K swept in steps of 4 via fp32 WMMA.
// A frag: lane L holds row L%16, K offset 2*(L/16). D: M = r + 8*(L/16).
// ---------------------------------------------------------------------------
__global__ void __launch_bounds__(32)
k_gemm1(const float* __restrict__ x, const float* __restrict__ W,
        float* __restrict__ h) {
    const int lane = threadIdx.x;
    const int m0   = blockIdx.x * 16;
    const int n0   = blockIdx.y * 16;
    const int row  = lane & 15;
    const int koff = (lane >> 4) << 1;       // 0 or 2

    const float* xr = x + (m0 + row) * 96;
    v8f acc = {};
    for (int k = 0; k < 96; k += 4) {
        v2f a, b;
        a.x = xr[k + koff];
        a.y = xr[k + koff + 1];
        b.x = W[(k + koff)     * C3 + n0 + row];
        b.y = W[(k + koff + 1) * C3 + n0 + row];
        acc = WMMA_F32_16x16x4(a, b, acc);
    }
    const int mbase = m0 + ((lane >> 4) << 3);
#pragma unroll
    for (int r = 0; r < 8; ++r)
        h[(mbase + r) * C3 + n0 + row] = acc[r];
}

// ---------------------------------------------------------------------------
// Kernel 5: edge scatter. One thread per (edge, 4-channel group):
//   features[dst, c:c+4] += h[src, c:c+4] * dinv[src]*dinv[dst]
// h and features are L2-resident (57.6 MB each, 192 MB L2).
// ---------------------------------------------------------------------------
__global__ void k_scatter(const float* __restrict__ h, const float* __restrict__ dinv,
                          const int* __restrict__ ei, float* __restrict__ feat) {
    const long long tid   = (long long)blockIdx.x * blockDim.x + threadIdx.x;
    const long long total = (long long)N_EDGES * (C3 / 4);
    if (tid >= total) return;
    const int e = (int)(tid / (C3 / 4));
    const int c = (int)(tid % (C3 / 4)) * 4;
    const int s = ei[e];
    const int d = ei[N_EDGES + e];
    const float norm = dinv[s] * dinv[d];
    const float4 hv = *reinterpret_cast<const float4*>(h + (long long)s * C3 + c);
    float* fp = feat + (long long)d * C3 + c;
    atomicAdd(fp + 0, hv.x * norm);
    atomicAdd(fp + 1, hv.y * norm);
    atomicAdd(fp + 2, hv.z * norm);
    atomicAdd(fp + 3, hv.w * norm);
}

// ---------------------------------------------------------------------------
// Kernel 6: fused update MLP + self-loop + residual, one wave per 16 nodes.
//   F   = features + h * dinv^2            (self-loop fused into the read)
//   T   = relu(F @ W1 + b1)                (16x32, staged through LDS)
//   out = x + T @ W2 + b2                  (16x96)
// ---------------------------------------------------------------------------
__global__ void __launch_bounds__(32)
k_update(const float* __restrict__ x,    const float* __restrict__ h,
         const float* __restrict__ feat, const float* __restrict__ dinv,
         const float* __restrict__ W1,   const float* __restrict__ b1,
         const float* __restrict__ W2,   const float* __restrict__ b2,
         float* __restrict__ out) {
    __shared__ float T[16 * HID];
    const int lane = threadIdx.x;
    const int m0   = blockIdx.x * 16;
    const int row  = lane & 15;
    const int hi   = lane >> 4;              // 0 or 1
    const int koff = hi << 1;                // 0 or 2
    const int node = m0 + row;

    const float dv = dinv[node];
    const float d2 = dv * dv;
    const float* fr = feat + node * C3;
    const float* hr = h    + node * C3;

    // ---- GEMM: F(16x288) @ W1(288x32) -> two 16x16 accumulators ----
    v8f acc0 = {}, acc1 = {};
    for (int k = 0; k < C3; k += 4) {
        v2f a, bA, bB;
        a.x  = fr[k + koff]     + hr[k + koff]     * d2;
        a.y  = fr[k + koff + 1] + hr[k + koff + 1] * d2;
        bA.x = W1[(k + koff)     * HID + row];
        bA.y = W1[(k + koff + 1) * HID + row];
        bB.x = W1[(k + koff)     * HID + 16 + row];
        bB.y = W1[(k + koff + 1) * HID + 16 + row];
        acc0 = WMMA_F32_16x16x4(a, bA, acc0);
        acc1 = WMMA_F32_16x16x4(a, bB, acc1);
    }

    // ---- bias + relu, restage through LDS (C-layout -> A-layout) ----
    const int   mb = hi << 3;
    const float bias0 = b1[row];
    const float bias1 = b1[16 + row];
#pragma unroll
    for (int r = 0; r < 8; ++r) {
        const float t0 = acc0[r] + bias0;
        const float t1 = acc1[r] + bias1;
        T[(mb + r) * HID + row]      = t0 > 0.f ? t0 : 0.f;
        T[(mb + r) * HID + 16 + row] = t1 > 0.f ? t1 : 0.f;
    }
    __syncthreads();

    // ---- GEMM: T(16x32) @ W2(32x96) -> six 16x16 accumulators ----
    v8f acc2[6];
#pragma unroll
    for (int t = 0; t < 6; ++t) acc2[t] = (v8f){};
    for (int k = 0; k < HID; k += 4) {
        v2f a;
        a.x = T[row * HID + k + koff];
        a.y = T[row * HID + k + koff + 1];
#pragma unroll
        for (int t = 0; t < 6; ++t) {
            v2f b;
            b.x = W2[(k + koff)     * 96 + t * 16 + row];
            b.y = W2[(k + koff + 1) * 96 + t * 16 + row];
            acc2[t] = WMMA_F32_16x16x4(a, b, acc2[t]);
        }
    }

    // ---- residual + b2, write out ----
#pragma unroll
    for (int t = 0; t < 6; ++t) {
        const int   col = t * 16 + row;
        const float bb  = b2[col];
#pragma unroll
        for (int r = 0; r < 8; ++r) {
            const int m = m0 + mb + r;
            out[m * 96 + col] = x[m * 96 + col] + acc2[t][r] + bb;
        }
    }
}

// ---------------------------------------------------------------------------
// Launch
// ---------------------------------------------------------------------------
extern "C" void kernel_launch(void* const* d_in, const int* in_sizes, int n_in,
                              void* d_out, int out_size, void* d_ws, size_t ws_size,
                              hipStream_t stream) {
    const float* x     = (const float*)d_in[0];
    const float* W_gcn = (const float*)d_in[1];
    const float* W1    = (const float*)d_in[2];
    const float* b1    = (const float*)d_in[3];
    const float* W2    = (const float*)d_in[4];
    const float* b2    = (const float*)d_in[5];
    const int*   ei    = (const int*)d_in[6];
    float*       out   = (float*)d_out;

    // workspace layout: dinv (50000 f32) | h (50000x288 f32) | features (50000x288 f32)
    char* ws = (char*)d_ws;
    float* dinv = (float*)ws;
    float* h    = (float*)(ws + 256 * ((N_NODES * 4 + 255) / 256));
    float* feat = h + (size_t)N_NODES * C3;

    // 1. init deg=1, features=0
    {
        const int n = N_NODES * C3 / 4;   // covers N_NODES too
        k_init<<<(n + 255) / 256, 256, 0, stream>>>(dinv, (float4*)feat);
    }
    // 2. in-degree atomics
    k_degree<<<(N_EDGES + 255) / 256, 256, 0, stream>>>(ei, dinv);
    // 3. rsqrt
    k_dinv<<<(N_NODES + 255) / 256, 256, 0, stream>>>(dinv);
    // 4. h = x @ W_gcn  (3125 x 18 tiles of 16x16, one wave each)
    k_gemm1<<<dim3(N_NODES / 16, C3 / 16), 32, 0, stream>>>(x, W_gcn, h);
    // 5. edge scatter-add
    {
        const long long total = (long long)N_EDGES * (C3 / 4);
        k_scatter<<<(int)((total + 255) / 256), 256, 0, stream>>>(h, dinv, ei, feat);
    }
    // 6. fused MLP + self-loop + residual
    k_update<<<N_NODES / 16, 32, 0, stream>>>(x, h, feat, dinv, W1, b1, W2, b2, out);
}